// SubspaceOT_5239860101134
// MI455X (gfx1250) — compile-verified
//
#include <hip/hip_runtime.h>

// SubspaceOT for MI455X (gfx1250): all GEMMs via V_WMMA_F32_16X16X4_F32.
// z kept transposed (N x K) exploiting symmetry of S, so X_c needs no transpose.
// Round 3: fix async-copy builtin pointer types per hipcc diagnostic
// (param 0 is 'int __attribute__((vector_size(16))) __device__ *').

typedef __attribute__((ext_vector_type(2))) float v2f;
typedef __attribute__((ext_vector_type(8))) float v8f;

#define N_FEAT 32768
#define D_DIM  512
#define K_DIM  512
#define H_DIM  256

#if __has_builtin(__builtin_amdgcn_global_load_async_to_lds_b128)
#define USE_ASYNC_COPY 1
typedef int i4vec __attribute__((vector_size(16)));
typedef __attribute__((address_space(1))) i4vec* gbl_i4_ptr;   // global (AS1)
typedef __attribute__((address_space(3))) i4vec* lds_i4_ptr;   // LDS (AS3)
#else
#define USE_ASYNC_COPY 0
#endif

__device__ __forceinline__ void async_drain()
{
#if USE_ASYNC_COPY && __has_builtin(__builtin_amdgcn_s_wait_asynccnt)
    __builtin_amdgcn_s_wait_asynccnt(0);
#endif
}

enum { EPI_NONE = 0, EPI_SCALE = 1, EPI_S = 2, EPI_BIAS_RELU = 3, EPI_LASSO = 4 };

// C[M x Ncols] = A[M x Kd] (row-major, lda) * B[Kd x Ncols] (row-major, ldb)
// Block tile 128x128, 256 threads (8 waves, 4x2 wave grid), 2x4 WMMA tiles/wave.
// As: stride 36 (conflict-free b64 frag loads).
// Bs: K-paired layout, pair-row stride 288: element (k,c) at [(k>>1)*288 + c*2 + (k&1)]
//     -> B fragment {B[k][c], B[k+1][c]} is one aligned b64; lane-halves hit
//        disjoint bank sets (288 mod 64 == 32).
template <int EPI>
__global__ __launch_bounds__(256) void gemm128_wmma(
    const float* __restrict__ A, const float* __restrict__ B, float* __restrict__ C,
    int Kd, int lda, int ldb, int ldc,
    const float* __restrict__ bias, const float* __restrict__ addmat,
    const float* __restrict__ lvec, const float* __restrict__ cptr)
{
    __shared__ float As[2][128 * 36];
    __shared__ float Bs[2][16 * 288];

    const int t    = threadIdx.x;
    const int lane = t & 31;
    const int wave = t >> 5;
    const int wm   = (wave & 3) * 32;   // 2 m-subtiles
    const int wn   = (wave >> 2) * 64;  // 4 n-subtiles
    const int lr   = lane & 15;
    const int lh   = lane >> 4;         // K 0/1 vs 2/3 (A,B); M 0-7 vs 8-15 (C)
    const int rowTile = blockIdx.y * 128;
    const int colTile = blockIdx.x * 128;

    auto stageA = [&](int buf, int kb) {
#pragma unroll
        for (int i = 0; i < 4; ++i) {
            int idx = i * 256 + t;           // 1024 float4
            int row = idx >> 3;
            int c4  = (idx & 7) << 2;
            const float* g = A + (size_t)(rowTile + row) * lda + kb + c4;
            float* l = &As[buf][row * 36 + c4];
#if USE_ASYNC_COPY
            __builtin_amdgcn_global_load_async_to_lds_b128(
                (gbl_i4_ptr)(unsigned long long)(size_t)g,
                (lds_i4_ptr)(unsigned int)(size_t)l,
                0, 0);
#else
            *(float4*)l = *(const float4*)g;
#endif
        }
    };

    auto stageB = [&](int buf, int kb) {
#pragma unroll
        for (int i = 0; i < 4; ++i) {
            int unit = i * 256 + t;          // 1024 units = 16 pair-rows x 64 col-pairs
            int p  = unit >> 6;              // pair-row (k = 2p, 2p+1)
            int cp = unit & 63;              // column pair (cols 2cp, 2cp+1)
            const float* g0 = B + (size_t)(kb + 2 * p)     * ldb + colTile + 2 * cp;
            const float* g1 = B + (size_t)(kb + 2 * p + 1) * ldb + colTile + 2 * cp;
            float2 r0 = *(const float2*)g0;
            float2 r1 = *(const float2*)g1;
            float4 w;
            w.x = r0.x; w.y = r1.x; w.z = r0.y; w.w = r1.y;  // interleave k-pair
            *(float4*)&Bs[buf][p * 288 + cp * 4] = w;
        }
    };

    v8f acc[2][4];
#pragma unroll
    for (int i = 0; i < 2; ++i)
#pragma unroll
        for (int j = 0; j < 4; ++j)
            acc[i][j] = (v8f)0.0f;

    stageA(0, 0);
    stageB(0, 0);
    async_drain();
    __syncthreads();

    const int nChunk = Kd >> 5;
    for (int ch = 0; ch < nChunk; ++ch) {
        const int cur = ch & 1;
        if (ch + 1 < nChunk) {
            stageA(cur ^ 1, (ch + 1) * 32);
            stageB(cur ^ 1, (ch + 1) * 32);
        }
        const float* Asb = As[cur];
        const float* Bsb = Bs[cur];
#pragma unroll
        for (int kk = 0; kk < 32; kk += 4) {
            const int kbase = kk + lh * 2;       // even
            v2f af[2], bf[4];
#pragma unroll
            for (int i = 0; i < 2; ++i)
                af[i] = *(const v2f*)&Asb[(wm + i * 16 + lr) * 36 + kbase];
#pragma unroll
            for (int j = 0; j < 4; ++j)
                bf[j] = *(const v2f*)&Bsb[(kbase >> 1) * 288 + (wn + j * 16 + lr) * 2];
#pragma unroll
            for (int i = 0; i < 2; ++i)
#pragma unroll
                for (int j = 0; j < 4; ++j)
                    acc[i][j] = __builtin_amdgcn_wmma_f32_16x16x4_f32(
                        false, af[i], false, bf[j], (short)0, acc[i][j], false, false);
        }
        async_drain();
        __syncthreads();
    }

    float inv_c = 1.0f;
    if constexpr (EPI == EPI_SCALE || EPI == EPI_S) inv_c = 1.0f / cptr[0];

#pragma unroll
    for (int i = 0; i < 2; ++i) {
#pragma unroll
        for (int j = 0; j < 4; ++j) {
            int gcol = colTile + wn + j * 16 + lr;
#pragma unroll
            for (int r = 0; r < 8; ++r) {
                int grow = rowTile + wm + i * 16 + lh * 8 + r;
                float v = acc[i][j][r];
                if constexpr (EPI == EPI_SCALE) {
                    v *= inv_c;
                } else if constexpr (EPI == EPI_S) {
                    v = (grow == gcol ? 1.0f : 0.0f) - v * inv_c;
                } else if constexpr (EPI == EPI_BIAS_RELU) {
                    v = fmaxf(v + bias[gcol], 0.0f);
                } else if constexpr (EPI == EPI_LASSO) {
                    v = fmaxf(v + addmat[(size_t)grow * ldc + gcol] - lvec[grow], 0.0f);
                }
                C[(size_t)grow * ldc + gcol] = v;
            }
        }
    }
}

// pos_basis = relu(basis) stored both (D,K) and transposed (K,D)
__global__ void relu_basis_kernel(const float* __restrict__ basis,
                                  float* __restrict__ PB, float* __restrict__ PBT)
{
    int idx = blockIdx.x * 256 + threadIdx.x;   // over D*K
    int d = idx / K_DIM, k = idx % K_DIM;
    float v = fmaxf(basis[idx], 0.0f);
    PB[idx] = v;
    PBT[(size_t)k * D_DIM + d] = v;
}

__global__ void transpose_w1_kernel(const float* __restrict__ W1, float* __restrict__ W1T)
{
    int idx = blockIdx.x * 256 + threadIdx.x;   // over H*D
    int h = idx / D_DIM, d = idx % D_DIM;
    W1T[(size_t)d * H_DIM + h] = W1[idx];
}

// l[n] = relu(h[n,:].W2 + b2) / c  — one wave32 per row
__global__ void lamb_kernel(const float* __restrict__ h, const float* __restrict__ W2,
                            const float* __restrict__ b2, const float* __restrict__ cptr,
                            float* __restrict__ l)
{
    int row  = (blockIdx.x * 256 + threadIdx.x) >> 5;
    int lane = threadIdx.x & 31;
    const float* hr = h + (size_t)row * H_DIM;
    float s = 0.0f;
#pragma unroll
    for (int j = 0; j < H_DIM / 32; ++j)
        s += hr[lane + j * 32] * W2[lane + j * 32];
#pragma unroll
    for (int off = 16; off; off >>= 1)
        s += __shfl_xor(s, off, 32);
    if (lane == 0)
        l[row] = fmaxf(s + b2[0], 0.0f) / cptr[0];
}

// z1 = relu(W - l)   (first lasso step with z0 = 0)
__global__ void zinit_kernel(const float* __restrict__ Wm, const float* __restrict__ lvec,
                             float* __restrict__ z)
{
    size_t idx = (size_t)blockIdx.x * 256 + threadIdx.x;   // over N*K
    int row = (int)(idx >> 9);                              // K_DIM == 512
    z[idx] = fmaxf(Wm[idx] - lvec[row], 0.0f);
}

extern "C" void kernel_launch(void* const* d_in, const int* in_sizes, int n_in,
                              void* d_out, int out_size, void* d_ws, size_t ws_size,
                              hipStream_t stream)
{
    const float* X     = (const float*)d_in[0];  // (N,D)
    const float* basis = (const float*)d_in[1];  // (D,K)
    const float* W1    = (const float*)d_in[2];  // (H,D)
    const float* b1    = (const float*)d_in[3];  // (H)
    const float* W2    = (const float*)d_in[4];  // (1,H)
    const float* b2    = (const float*)d_in[5];  // (1)
    const float* cptr  = (const float*)d_in[6];  // scalar ||basis||^2

    // Workspace layout (floats); total ~238 MB
    float* ws   = (float*)d_ws;
    float* PB   = ws;                                    // D*K
    float* PBT  = PB  + (size_t)D_DIM * K_DIM;           // K*D
    float* W1T  = PBT + (size_t)K_DIM * D_DIM;           // D*H
    float* Smat = W1T + (size_t)D_DIM * H_DIM;           // K*K
    float* lv   = Smat + (size_t)K_DIM * K_DIM;          // N
    float* hbuf = lv   + N_FEAT;                         // N*H
    float* Wm   = hbuf + (size_t)N_FEAT * H_DIM;         // N*K  (= DTy_c^T)
    float* zA   = Wm   + (size_t)N_FEAT * K_DIM;         // N*K
    float* zB   = zA   + (size_t)N_FEAT * K_DIM;         // N*K

    float* Xc = (float*)d_out;                           // (N,K)
    float* Xp = Xc + (size_t)N_FEAT * K_DIM;             // (N,D)

    relu_basis_kernel<<<D_DIM * K_DIM / 256, 256, 0, stream>>>(basis, PB, PBT);
    transpose_w1_kernel<<<H_DIM * D_DIM / 256, 256, 0, stream>>>(W1, W1T);

    // S = I - PBT(KxD) * PB(DxK) / c
    gemm128_wmma<EPI_S><<<dim3(K_DIM / 128, K_DIM / 128), 256, 0, stream>>>(
        PBT, PB, Smat, D_DIM, D_DIM, K_DIM, K_DIM, nullptr, nullptr, nullptr, cptr);

    // h = relu(X(NxD) * W1T(DxH) + b1)
    gemm128_wmma<EPI_BIAS_RELU><<<dim3(H_DIM / 128, N_FEAT / 128), 256, 0, stream>>>(
        X, W1T, hbuf, D_DIM, D_DIM, H_DIM, H_DIM, b1, nullptr, nullptr, nullptr);

    lamb_kernel<<<N_FEAT * 32 / 256, 256, 0, stream>>>(hbuf, W2, b2, cptr, lv);

    // Wm = X(NxD) * PB(DxK) / c   (== DTy_c transposed)
    gemm128_wmma<EPI_SCALE><<<dim3(K_DIM / 128, N_FEAT / 128), 256, 0, stream>>>(
        X, PB, Wm, D_DIM, D_DIM, K_DIM, K_DIM, nullptr, nullptr, nullptr, cptr);

    // Iter 1: z1 = relu(Wm - l)
    zinit_kernel<<<(int)((size_t)N_FEAT * K_DIM / 256), 256, 0, stream>>>(Wm, lv, zA);

    // Iters 2..10: z' = relu(z * S + Wm - l)   (S symmetric -> transposed form valid)
    for (int it = 2; it <= 10; ++it) {
        float* src = (it & 1) ? zB : zA;
        float* dst = (it == 10) ? Xc : ((it & 1) ? zA : zB);
        gemm128_wmma<EPI_LASSO><<<dim3(K_DIM / 128, N_FEAT / 128), 256, 0, stream>>>(
            src, Smat, dst, K_DIM, K_DIM, K_DIM, K_DIM, nullptr, Wm, lv, nullptr);
    }

    // X_p = X_c(NxK) * PBT(KxD)
    gemm128_wmma<EPI_NONE><<<dim3(D_DIM / 128, N_FEAT / 128), 256, 0, stream>>>(
        Xc, PBT, Xp, K_DIM, K_DIM, D_DIM, D_DIM, nullptr, nullptr, nullptr, nullptr);
}